// KVCacheOps_19353122635895
// MI455X (gfx1250) — compile-verified
//
#include <hip/hip_runtime.h>

// ---------------------------------------------------------------------------
// Reference semantics: out[kv][s][h][d] = new_data[s][h][d] for kv in {0,1}.
// The (page, layer) slot is overwritten then read back, so d_out depends only
// on new_data. Pure data movement:
//   read 51,200 f32 (204.8 KB) + write 102,400 f32 (409.6 KB) ~= 614 KB.
// At 23.3 TB/s HBM that is ~26 ns of DRAM time -> launch-latency bound.
// Optimal shape: one tiny kernel, B128 transfers, exact-fit grid, minimal
// instruction count. We use the gfx1250 async global<->LDS data-mover path
// (GLOBAL_LOAD_ASYNC_TO_LDS_B128 / GLOBAL_STORE_ASYNC_FROM_LDS_B128 +
// S_WAIT_ASYNCCNT) so the copy bypasses VGPRs. Each lane stores from the same
// LDS slot it loaded, so only the per-wave asynccnt wait is needed (no
// workgroup barrier); outstanding stores drain at S_ENDPGM's implicit
// wait-idle.
// ---------------------------------------------------------------------------

#define SRC_FLOATS (16 * 32 * 100)     // 51200 floats in new_data [S,H,D]
#define SRC_VEC4   (SRC_FLOATS / 4)    // 12800 16B vectors
#define BLOCK      128
#define GRID       (SRC_VEC4 / BLOCK)  // 100 blocks, exact fit, no tail

typedef int   v4i __attribute__((ext_vector_type(4)));
typedef float f4  __attribute__((ext_vector_type(4)));

#if defined(__AMDGCN__) &&                                                    \
    __has_builtin(__builtin_amdgcn_global_load_async_to_lds_b128) &&          \
    __has_builtin(__builtin_amdgcn_global_store_async_from_lds_b128)
#define KV_HAVE_ASYNC_LDS 1
// Builtin signature (from clang diagnostic): first param is
// 'v4i __device__*' == v4i addrspace(1)*; LDS param is v4i addrspace(3)*.
typedef __attribute__((address_space(1))) v4i gv4i;
typedef __attribute__((address_space(3))) v4i sv4i;
#endif

__global__ __launch_bounds__(BLOCK) void kv_bcast_copy(const f4* __restrict__ src,
                                                       f4* __restrict__ dst) {
    const int gi = (int)blockIdx.x * BLOCK + (int)threadIdx.x;  // exact coverage
#ifdef KV_HAVE_ASYNC_LDS
    __shared__ v4i tile[BLOCK];
    sv4i* slot = (sv4i*)&tile[threadIdx.x];

    // HBM -> LDS, 16B per lane, async (no VGPR data path)
    __builtin_amdgcn_global_load_async_to_lds_b128((gv4i*)(src + gi), slot,
                                                   /*offset=*/0, /*cpol=*/0);

    // Wave-local ordering: async LDS writes may complete out of order, so
    // drain this wave's async ops before reading LDS back out.
#if __has_builtin(__builtin_amdgcn_s_wait_asynccnt)
    __builtin_amdgcn_s_wait_asynccnt(0);
#else
    asm volatile("s_wait_asynccnt 0" ::: "memory");
#endif

    // LDS -> HBM, once per KV plane (broadcast over KV dim)
    __builtin_amdgcn_global_store_async_from_lds_b128((gv4i*)(dst + gi), slot,
                                                      /*offset=*/0, /*cpol=*/0);
    __builtin_amdgcn_global_store_async_from_lds_b128(
        (gv4i*)(dst + SRC_VEC4 + gi), slot, /*offset=*/0, /*cpol=*/0);
    // S_ENDPGM performs an implicit wait-idle; ASYNCcnt reaches 0 before the
    // wave terminates, so the LDS allocation stays live for the stores.
#else
    // Fallback (toolchain without async-LDS builtins): plain B128 copy.
    const f4 v = src[gi];
    dst[gi] = v;
    dst[SRC_VEC4 + gi] = v;
#endif
}

extern "C" void kernel_launch(void* const* d_in, const int* in_sizes, int n_in,
                              void* d_out, int out_size, void* d_ws, size_t ws_size,
                              hipStream_t stream) {
    (void)in_sizes; (void)n_in; (void)d_ws; (void)ws_size; (void)out_size;
    // d_in[0] = kvcache (unused: output is independent of it, and inputs must
    //           not be mutated), d_in[2]/d_in[3] = page/layer indices (unused).
    const f4* new_data = (const f4*)d_in[1];  // [16,32,100] f32, contiguous
    f4*       out      = (f4*)d_out;          // [2,16,32,100] f32
    kv_bcast_copy<<<GRID, BLOCK, 0, stream>>>(new_data, out);
}